// TopKPixelCrossEntropyLoss_42520176230819
// MI455X (gfx1250) — compile-verified
//
#include <hip/hip_runtime.h>
#include <hip/hip_bf16.h>
#include <stdint.h>

// Problem constants (from reference): B=16, C=4, H=W=512, K_FRAC=0.1
#define NSLICE 64                 // B*C slices
#define SLICE_ELEMS 262144        // H*W
#define KSEL 26214u               // int(0.1 * 262144)
#define NB1 4096                  // coarse bins: float bits [30:19]
#define NB2 4096                  // fine bins:   float bits [18:7]
#define NB3 128                   // last bins:   float bits [6:0]
#define BLKS_PER_SLICE 16
#define TPB 256
#define ELEMS_PER_BLOCK (SLICE_ELEMS / BLKS_PER_SLICE)   // 16384
#define TILE 1024                 // floats per block-iteration (256 thr * 4)
#define ITERS (ELEMS_PER_BLOCK / TILE)                   // 16

// Workspace layout (uint32 words). Total ~3.0 MB.
#define OFF_H1 0                                   // coarse counts
#define OFF_H2 (OFF_H1 + NSLICE * NB1)             // fine counts (within b1)
#define OFF_S2 (OFF_H2 + NSLICE * NB2)             // fine per-bin float sums (within b1)
#define OFF_H3 (OFF_S2 + NSLICE * NB2)             // last counts (within b1,f2)
#define OFF_ST (OFF_H3 + NSLICE * NB3)
#define ST_WORDS 8
// state/slice: [0]=b1 [1]=cnt_gt1 [2]=f2 [3]=cnt_gt2 [6]=running top-k sum (f32 bits)
#define WS_WORDS (OFF_ST + NSLICE * ST_WORDS)

// ---------------- CDNA5 async global->LDS staging (gfx1250) ----------------
__device__ __forceinline__ void async_b128_to_lds(unsigned lds_off, const void* gaddr) {
    // GV mode: per-lane copy of 16 bytes, tracked by ASYNCcnt.
    asm volatile("global_load_async_to_lds_b128 %0, %1, off"
                 :: "v"(lds_off), "v"((unsigned long long)(uintptr_t)gaddr)
                 : "memory");
}
template <int N>
__device__ __forceinline__ void wait_async_le() {
    if constexpr (N == 0) asm volatile("s_wait_asynccnt 0x0" ::: "memory");
    else                  asm volatile("s_wait_asynccnt 0x2" ::: "memory");
}
__device__ __forceinline__ unsigned lds_offset(const void* p) {
    // Generic LDS pointer = {shared aperture hi, 32-bit LDS offset}; truncate.
    return (unsigned)(uintptr_t)p;
}

// Stable BCE-with-logits, loss >= 0 always (so IEEE bits are order-preserving).
__device__ __forceinline__ float bce(float x, float y) {
    return fmaxf(x, 0.0f) - x * y + log1pf(__expf(-fabsf(x)));
}

// ---------------- K0: zero workspace ----------------
__global__ void k_zero(unsigned* __restrict__ ws) {
    for (int i = blockIdx.x * blockDim.x + threadIdx.x; i < WS_WORDS;
         i += gridDim.x * blockDim.x)
        ws[i] = 0u;
}

// ============ Pass 1: coarse histogram on bits[30:19] ============
__global__ __launch_bounds__(TPB) void k_hist1(const float* __restrict__ logits,
                                               const float* __restrict__ targets,
                                               unsigned* __restrict__ ws) {
    __shared__ unsigned lhist[NB1];
    __shared__ __align__(16) float sL[2][TILE];
    __shared__ __align__(16) float sT[2][TILE];
    const int tid   = threadIdx.x;
    const int slice = blockIdx.x / BLKS_PER_SLICE;
    const int base  = slice * SLICE_ELEMS + (blockIdx.x % BLKS_PER_SLICE) * ELEMS_PER_BLOCK;
    for (int i = tid; i < NB1; i += TPB) lhist[i] = 0u;
    __syncthreads();
    const unsigned ldsL[2] = { lds_offset(&sL[0][tid * 4]), lds_offset(&sL[1][tid * 4]) };
    const unsigned ldsT[2] = { lds_offset(&sT[0][tid * 4]), lds_offset(&sT[1][tid * 4]) };
    // prologue: tile 0 -> buffer 0
    async_b128_to_lds(ldsL[0], logits + base + tid * 4);
    async_b128_to_lds(ldsT[0], targets + base + tid * 4);
    for (int it = 0; it < ITERS; ++it) {
        if (it + 1 < ITERS) {
            const int nidx = base + (it + 1) * TILE + tid * 4;
            async_b128_to_lds(ldsL[(it + 1) & 1], logits + nidx);
            async_b128_to_lds(ldsT[(it + 1) & 1], targets + nidx);
            wait_async_le<2>();      // tile `it` has landed
        } else {
            wait_async_le<0>();
        }
        const float* bl = sL[it & 1];
        const float* bt = sT[it & 1];
#pragma unroll
        for (int j = 0; j < 4; ++j) {
            float v = bce(bl[tid * 4 + j], bt[tid * 4 + j]);
            atomicAdd(&lhist[(__float_as_uint(v) >> 19) & (NB1 - 1)], 1u);
        }
    }
    __syncthreads();
    for (int i = tid; i < NB1; i += TPB) {
        unsigned c = lhist[i];
        if (c) atomicAdd(&ws[OFF_H1 + slice * NB1 + i], c);
    }
}

// ---------------- suffix-scan helper for 4096-bin histograms ----------------
__device__ void scan4096(const unsigned* __restrict__ h, unsigned target,
                         unsigned* out_bin, unsigned* out_cnt_above) {
    __shared__ unsigned csum[TPB];
    const int tid = threadIdx.x;
    unsigned mysum = 0;
#pragma unroll
    for (int j = 0; j < NB1 / TPB; ++j) mysum += h[tid * (NB1 / TPB) + j];
    csum[tid] = mysum;
    __syncthreads();
    if (tid == 0) {
        unsigned cum = 0;
        int ci = TPB - 1;
        for (; ci > 0; --ci) {
            if (cum + csum[ci] >= target) break;
            cum += csum[ci];
        }
        unsigned bin = 0;
        for (int j = NB1 / TPB - 1; j >= 0; --j) {
            unsigned c = h[ci * (NB1 / TPB) + j];
            if (cum + c >= target) { bin = (unsigned)(ci * (NB1 / TPB) + j); break; }
            cum += c;
        }
        *out_bin = bin;
        *out_cnt_above = cum;   // count strictly above chosen bin
    }
    __syncthreads();
}

__global__ __launch_bounds__(TPB) void k_scan1(unsigned* __restrict__ ws) {
    const int s = blockIdx.x;
    __shared__ unsigned rbin, rcnt;
    scan4096(ws + OFF_H1 + s * NB1, KSEL, &rbin, &rcnt);
    if (threadIdx.x == 0) {
        unsigned* st = ws + OFF_ST + s * ST_WORDS;
        st[0] = rbin;
        st[1] = rcnt;
    }
}

// ============ Pass 2: fine hist within b1 + sums ============
// Accumulates: fine counts, per-fine-bin value sums, and the per-slice sum of
// all values strictly above coarse bin b1 (into st[6]).
__global__ __launch_bounds__(TPB) void k_hist2(const float* __restrict__ logits,
                                               const float* __restrict__ targets,
                                               unsigned* __restrict__ ws) {
    __shared__ unsigned lcnt[NB2];
    __shared__ float    lsum[NB2];
    __shared__ __align__(16) float sL[2][TILE];
    __shared__ __align__(16) float sT[2][TILE];
    __shared__ float red[TPB];
    const int tid   = threadIdx.x;
    const int slice = blockIdx.x / BLKS_PER_SLICE;
    const int base  = slice * SLICE_ELEMS + (blockIdx.x % BLKS_PER_SLICE) * ELEMS_PER_BLOCK;
    const unsigned b1 = ws[OFF_ST + slice * ST_WORDS + 0];
    for (int i = tid; i < NB2; i += TPB) { lcnt[i] = 0u; lsum[i] = 0.0f; }
    __syncthreads();
    const unsigned ldsL[2] = { lds_offset(&sL[0][tid * 4]), lds_offset(&sL[1][tid * 4]) };
    const unsigned ldsT[2] = { lds_offset(&sT[0][tid * 4]), lds_offset(&sT[1][tid * 4]) };
    async_b128_to_lds(ldsL[0], logits + base + tid * 4);
    async_b128_to_lds(ldsT[0], targets + base + tid * 4);
    float acc = 0.0f;   // sum of values with coarse bin > b1
    for (int it = 0; it < ITERS; ++it) {
        if (it + 1 < ITERS) {
            const int nidx = base + (it + 1) * TILE + tid * 4;
            async_b128_to_lds(ldsL[(it + 1) & 1], logits + nidx);
            async_b128_to_lds(ldsT[(it + 1) & 1], targets + nidx);
            wait_async_le<2>();
        } else {
            wait_async_le<0>();
        }
        const float* bl = sL[it & 1];
        const float* bt = sT[it & 1];
#pragma unroll
        for (int j = 0; j < 4; ++j) {
            float v = bce(bl[tid * 4 + j], bt[tid * 4 + j]);
            unsigned u  = __float_as_uint(v);
            unsigned cb = u >> 19;
            if (cb > b1) {
                acc += v;
            } else if (cb == b1) {
                unsigned fb = (u >> 7) & (NB2 - 1);
                atomicAdd(&lcnt[fb], 1u);
                atomicAdd(&lsum[fb], v);
            }
        }
    }
    __syncthreads();
    for (int i = tid; i < NB2; i += TPB) {
        unsigned c = lcnt[i];
        if (c) {
            atomicAdd(&ws[OFF_H2 + slice * NB2 + i], c);
            atomicAdd((float*)&ws[OFF_S2 + slice * NB2 + i], lsum[i]);
        }
    }
    // block-reduce acc, one float atomic per block into st[6]
    red[tid] = acc;
    __syncthreads();
    for (int w = TPB / 2; w > 0; w >>= 1) {
        if (tid < w) red[tid] += red[tid + w];
        __syncthreads();
    }
    if (tid == 0)
        atomicAdd((float*)(ws + OFF_ST + slice * ST_WORDS + 6), red[0]);
}

// scan fine hist: pick f2, add sum of fine bins strictly above f2 into st[6]
__global__ __launch_bounds__(TPB) void k_scan2(unsigned* __restrict__ ws) {
    const int s   = blockIdx.x;
    const int tid = threadIdx.x;
    unsigned* st  = ws + OFF_ST + s * ST_WORDS;
    __shared__ unsigned rbin, rcnt;
    scan4096(ws + OFF_H2 + s * NB2, KSEL - st[1], &rbin, &rcnt);
    // parallel sum of fine-bin sums strictly above rbin
    __shared__ float fred[TPB];
    const float* fs = (const float*)(ws + OFF_S2 + s * NB2);
    const unsigned b = rbin;
    float a = 0.0f;
    for (int j = tid; j < NB2; j += TPB)
        if ((unsigned)j > b) a += fs[j];
    fred[tid] = a;
    __syncthreads();
    for (int w = TPB / 2; w > 0; w >>= 1) {
        if (tid < w) fred[tid] += fred[tid + w];
        __syncthreads();
    }
    if (tid == 0) {
        st[2] = rbin;
        st[3] = st[1] + rcnt;   // total strictly above (b1,f2)
        float cur = __uint_as_float(st[6]);
        st[6] = __float_as_uint(cur + fred[0]);
    }
}

// ============ Pass 3: last histogram on bits[6:0] within (b1,f2) ============
__global__ __launch_bounds__(TPB) void k_hist3(const float* __restrict__ logits,
                                               const float* __restrict__ targets,
                                               unsigned* __restrict__ ws) {
    __shared__ unsigned lhist[NB3];
    __shared__ __align__(16) float sL[2][TILE];
    __shared__ __align__(16) float sT[2][TILE];
    const int tid   = threadIdx.x;
    const int slice = blockIdx.x / BLKS_PER_SLICE;
    const int base  = slice * SLICE_ELEMS + (blockIdx.x % BLKS_PER_SLICE) * ELEMS_PER_BLOCK;
    const unsigned* st = ws + OFF_ST + slice * ST_WORDS;
    const unsigned pref = (st[0] << 12) | st[2];   // bits [30:7]
    for (int i = tid; i < NB3; i += TPB) lhist[i] = 0u;
    __syncthreads();
    const unsigned ldsL[2] = { lds_offset(&sL[0][tid * 4]), lds_offset(&sL[1][tid * 4]) };
    const unsigned ldsT[2] = { lds_offset(&sT[0][tid * 4]), lds_offset(&sT[1][tid * 4]) };
    async_b128_to_lds(ldsL[0], logits + base + tid * 4);
    async_b128_to_lds(ldsT[0], targets + base + tid * 4);
    for (int it = 0; it < ITERS; ++it) {
        if (it + 1 < ITERS) {
            const int nidx = base + (it + 1) * TILE + tid * 4;
            async_b128_to_lds(ldsL[(it + 1) & 1], logits + nidx);
            async_b128_to_lds(ldsT[(it + 1) & 1], targets + nidx);
            wait_async_le<2>();
        } else {
            wait_async_le<0>();
        }
        const float* bl = sL[it & 1];
        const float* bt = sT[it & 1];
#pragma unroll
        for (int j = 0; j < 4; ++j) {
            unsigned u = __float_as_uint(bce(bl[tid * 4 + j], bt[tid * 4 + j]));
            if ((u >> 7) == pref) atomicAdd(&lhist[u & (NB3 - 1)], 1u);
        }
    }
    __syncthreads();
    for (int i = tid; i < NB3; i += TPB) {
        unsigned c = lhist[i];
        if (c) atomicAdd(&ws[OFF_H3 + slice * NB3 + i], c);
    }
}

// finish: all values in one last-level bin share one exact bit pattern,
// so the (b1,f2)-bin contribution is a closed form -- no more data passes.
__global__ void k_scan3(unsigned* __restrict__ ws) {
    const int s = blockIdx.x;
    if (threadIdx.x != 0) return;
    unsigned* st = ws + OFF_ST + s * ST_WORDS;
    const unsigned* h = ws + OFF_H3 + s * NB3;
    const unsigned target = KSEL - st[3];          // >= 1
    unsigned cum = 0, g = 0;
    for (int j = NB3 - 1; j >= 0; --j) {
        unsigned c = h[j];
        if (cum + c >= target) { g = (unsigned)j; break; }
        cum += c;
    }
    const unsigned hi = (st[0] << 19) | (st[2] << 7);
    float part = 0.0f;
    for (unsigned j = NB3 - 1; j > g; --j)
        part += (float)h[j] * __uint_as_float(hi | j);  // identical-value bins
    part += (float)(target - cum) * __uint_as_float(hi | g);  // kth ties
    st[6] = __float_as_uint(__uint_as_float(st[6]) + part);
}

// ---------------- final combine ----------------
__global__ void k_final(const unsigned* __restrict__ ws, float* __restrict__ out) {
    __shared__ float part[NSLICE];
    const int tid = threadIdx.x;                   // 64 threads
    part[tid] = __uint_as_float(ws[OFF_ST + tid * ST_WORDS + 6]);
    __syncthreads();
    if (tid == 0) {
        float t = 0.0f;
        for (int i = 0; i < NSLICE; ++i) t += part[i];
        out[0] = t / (float)(NSLICE * KSEL);
    }
}

extern "C" void kernel_launch(void* const* d_in, const int* in_sizes, int n_in,
                              void* d_out, int out_size, void* d_ws, size_t ws_size,
                              hipStream_t stream) {
    const float* logits  = (const float*)d_in[0];
    const float* targets = (const float*)d_in[1];
    unsigned* ws  = (unsigned*)d_ws;   // needs WS_WORDS*4 ~= 3.0 MB
    float*    out = (float*)d_out;

    k_zero <<<512, TPB, 0, stream>>>(ws);
    k_hist1<<<NSLICE * BLKS_PER_SLICE, TPB, 0, stream>>>(logits, targets, ws);
    k_scan1<<<NSLICE, TPB, 0, stream>>>(ws);
    k_hist2<<<NSLICE * BLKS_PER_SLICE, TPB, 0, stream>>>(logits, targets, ws);
    k_scan2<<<NSLICE, TPB, 0, stream>>>(ws);
    k_hist3<<<NSLICE * BLKS_PER_SLICE, TPB, 0, stream>>>(logits, targets, ws);
    k_scan3<<<NSLICE, 64, 0, stream>>>(ws);
    k_final<<<1, NSLICE, 0, stream>>>(ws, out);
}